// RandomElasticDeformation3D_54812372631573
// MI455X (gfx1250) — compile-verified
//
#include <hip/hip_runtime.h>
#include <hip/hip_bf16.h>
#include <math.h>

// ---------------------------------------------------------------------------
// RandomElasticDeformation3D for MI455X (gfx1250)
//
// flow = ALPHA * (G*R) (x) (G*R) (x) (G*R) * (2*coarse-1)   (separable, linear)
//   K = G*R is a single 128x4 matrix shared by all three axes.
// Each workgroup builds the flow for one (b, d, h-half) 64x128 half-slice
// with V_WMMA_F32_16X16X4_F32 (M=64,N=128,K=4 per channel) directly in LDS
// (~107KB -> two workgroups per 320KB WGP), then runs the bandwidth-bound
// trilinear/nearest gather out of that LDS flow with zero HBM traffic for
// the flow field. Outputs + label use non-temporal hints so the 33.5MB image
// (re-read ~8x by the gather) stays resident in the 192MB L2.
// ---------------------------------------------------------------------------

typedef __attribute__((ext_vector_type(2))) float v2f;
typedef __attribute__((ext_vector_type(8))) float v8f;

#define ALPHA 35.0f
#define SIGMA 2.5f
#define DIMN  128
#define HROWS 64     // h-rows per workgroup (half slice)
#define CN    4      // coarse grid size per axis
#define GTAPS 16     // ksize-1 taps, ax = -7..8 (faithful to reference)
#define GPADLO 7     // SAME padding low side for even-length kernel

// Keys cubic kernel, a = -0.5 (jax.image.resize 'cubic')
__device__ __forceinline__ float keys_cubic(float x) {
    x = fabsf(x);
    if (x < 1.0f) return ((1.5f * x - 2.5f) * x) * x + 1.0f;
    if (x < 2.0f) return ((-0.5f * x + 2.5f) * x - 4.0f) * x + 2.0f;
    return 0.0f;
}

// ---------------------------------------------------------------------------
// Setup: K[128][4] = G (128x128 SAME-padded gaussian) @ R (128x4 cubic resize)
// One block of 128 threads; thread d computes row d of K. Output -> d_ws.
// ---------------------------------------------------------------------------
__global__ void build_K_kernel(float* __restrict__ Kmat) {
    const int d = threadIdx.x;
    if (d >= DIMN) return;

    float g[GTAPS];
    float gsum = 0.0f;
    #pragma unroll
    for (int t = 0; t < GTAPS; ++t) {
        const float ax = (float)(t - GPADLO);
        g[t] = __expf(-(ax * ax) / (2.0f * SIGMA * SIGMA));
        gsum += g[t];
    }
    const float ginv = 1.0f / gsum;

    float acc[CN] = {0.0f, 0.0f, 0.0f, 0.0f};
    #pragma unroll
    for (int t = 0; t < GTAPS; ++t) {
        const int n = d - GPADLO + t;           // zero padding outside [0,127]
        if (n < 0 || n >= DIMN) continue;
        const float sf = ((float)n + 0.5f) * ((float)CN / (float)DIMN) - 0.5f;
        float w[CN], ws = 0.0f;
        #pragma unroll
        for (int i = 0; i < CN; ++i) { w[i] = keys_cubic(sf - (float)i); ws += w[i]; }
        const float winv = 1.0f / ws;           // jax normalizes per output sample
        const float gt = g[t] * ginv;
        #pragma unroll
        for (int i = 0; i < CN; ++i) acc[i] += gt * w[i] * winv;
    }
    #pragma unroll
    for (int i = 0; i < CN; ++i) Kmat[d * CN + i] = acc[i];
}

// ---------------------------------------------------------------------------
// Fused flow-build (WMMA) + warp kernel. One workgroup per (b, d, h-half).
// blockDim = 512 (16 wave32s). Dynamic LDS layout (floats):
//   sflow [3][64*128]    24576
//   sK    [128*4]          512
//   sQ2   [3][4][4]         48
//   sT    [3][4][128]     1536      total 26672 floats = 106688 B
// ---------------------------------------------------------------------------
#define NTHREADS 512
#define HSLICE   (HROWS * DIMN)               // 8192 voxels per block
#define SMEM_FLOATS (3 * HSLICE + DIMN * CN + 3 * CN * CN + 3 * CN * DIMN)

__global__ __launch_bounds__(NTHREADS) void elastic_warp_kernel(
    const float* __restrict__ image,    // (B,128,128,128,2)
    const float* __restrict__ label,    // (B,128,128,128,1)
    const float* __restrict__ coarse,   // (B,4,4,4,3)
    const float* __restrict__ Kmat,     // (128,4) from build_K_kernel
    float* __restrict__ out_img,        // (B,128,128,128,2)
    float* __restrict__ out_lbl)        // (B,128,128,128,1)
{
    extern __shared__ float smem[];
    float* sflow = smem;                           // 3 * 8192
    float* sK    = sflow + 3 * HSLICE;             // 512
    float* sQ2   = sK + DIMN * CN;                 // 48
    float* sT    = sQ2 + 3 * CN * CN;              // 1536

    const int tid = threadIdx.x;
    const int bi  = blockIdx.x;                    // 0 .. 511
    const int b   = bi >> 8;
    const int rem0 = bi & 255;
    const int d   = rem0 >> 1;
    const int h0  = (rem0 & 1) * HROWS;            // 0 or 64

    // ---- stage 0: K matrix into LDS -------------------------------------
    if (tid < DIMN * CN) sK[tid] = Kmat[tid];
    __syncthreads();

    // ---- stage 1: Q2[c][j][k] = ALPHA * sum_i K[d,i]*(2*coarse-1) -------
    if (tid < 3 * CN * CN) {
        const int c = tid >> 4;
        const int r = tid & 15;
        const int j = r >> 2;
        const int k = r & 3;
        float q = 0.0f;
        #pragma unroll
        for (int i = 0; i < CN; ++i) {
            const float cf = coarse[((((b * CN + i) * CN + j) * CN + k) * 3) + c];
            q += sK[d * CN + i] * (2.0f * cf - 1.0f);
        }
        sQ2[(c * CN + j) * CN + k] = ALPHA * q;
    }
    __syncthreads();

    // ---- stage 2: T[c][j][w] = sum_k K[w,k] * Q2[c][j][k] ---------------
    for (int e = tid; e < 3 * CN * DIMN; e += NTHREADS) {
        const int w  = e & (DIMN - 1);
        const int cj = e >> 7;                     // c*4 + j
        float t = 0.0f;
        #pragma unroll
        for (int k = 0; k < CN; ++k)
            t += sK[w * CN + k] * sQ2[cj * CN + k];
        sT[cj * DIMN + w] = t;
    }
    __syncthreads();

    // ---- stage 3: flow_c (64x128) = K[h0:h0+64,:] (64x4) @ T_c (4x128) --
    // 3 channels * 4x8 tiles of 16x16 = 96 tiles; 6 per wave (uniform loop,
    // EXEC all ones as required by WMMA).
    {
        const int wid  = tid >> 5;
        const int lane = tid & 31;
        const int m    = lane & 15;                // row within A / col within B
        const int kb   = (lane >> 4) << 1;         // k-pair base: 0 or 2
        #pragma unroll
        for (int q = 0; q < 6; ++q) {
            const int tt  = wid * 6 + q;           // 0..95
            const int c   = tt >> 5;
            const int rem = tt & 31;
            const int mi  = rem >> 3;              // local h-tile (0..3)
            const int ni  = rem & 7;               // w-tile (0..7)
            // A: 16x4 f32 -> 2 VGPRs: lanes 0-15 K={0,1}, lanes 16-31 K={2,3}
            v2f a;
            a.x = sK[(h0 + mi * 16 + m) * CN + kb];
            a.y = sK[(h0 + mi * 16 + m) * CN + kb + 1];
            // B: 4x16 f32 -> 2 VGPRs: rows striped across lanes per VGPR
            v2f bm;
            bm.x = sT[(c * CN + kb)     * DIMN + ni * 16 + m];
            bm.y = sT[(c * CN + kb + 1) * DIMN + ni * 16 + m];
            v8f acc = {};
            acc = __builtin_amdgcn_wmma_f32_16x16x4_f32(
                false, a, false, bm, (short)0, acc, false, false);
            // D: VGPR v, lanes 0-15 -> row v, lanes 16-31 -> row v+8
            const int row0 = mi * 16 + ((lane >> 4) << 3);  // local row
            const int col  = ni * 16 + m;
            float* fp = sflow + c * HSLICE + col;
            #pragma unroll
            for (int v = 0; v < 8; ++v) fp[(row0 + v) * DIMN] = acc[v];
        }
    }
    __syncthreads();

    // ---- stage 4: gather (trilinear image, nearest label) ---------------
    const float* imgB = image + (size_t)b * DIMN * DIMN * DIMN * 2;
    const float* lblB = label + (size_t)b * DIMN * DIMN * DIMN;
    float* oImg = out_img + (((size_t)(b * DIMN + d) * DIMN + h0) * DIMN) * 2;
    float* oLbl = out_lbl + (((size_t)(b * DIMN + d) * DIMN + h0) * DIMN);

    for (int idx = tid; idx < HSLICE; idx += NTHREADS) {
        const int hl = idx >> 7;                   // local h row (0..63)
        const int h  = h0 + hl;
        const int w  = idx & (DIMN - 1);

        const float zd = (float)d + sflow[idx];
        const float zh = (float)h + sflow[HSLICE + idx];
        const float zw = (float)w + sflow[2 * HSLICE + idx];

        // trilinear, mode='constant' cval=0 (OOB corners contribute 0)
        const float fd = floorf(zd), fh = floorf(zh), fw = floorf(zw);
        const int z0 = (int)fd, y0 = (int)fh, x0 = (int)fw;
        const float td = zd - fd, th = zh - fh, tw = zw - fw;
        const float sd = 1.0f - td, sh = 1.0f - th, sw = 1.0f - tw;

        // pre-touch the likely cacheline for the gather footprint
        if ((unsigned)z0 < (unsigned)DIMN && (unsigned)y0 < (unsigned)DIMN &&
            (unsigned)x0 < (unsigned)DIMN) {
            __builtin_prefetch(imgB + (((z0 * DIMN + y0) * DIMN + x0) << 1), 0, 1);
        }

        float c0 = 0.0f, c1 = 0.0f;
        #pragma unroll
        for (int dz = 0; dz < 2; ++dz) {
            const int zz = z0 + dz;
            if ((unsigned)zz >= (unsigned)DIMN) continue;
            const float wz = dz ? td : sd;
            #pragma unroll
            for (int dy = 0; dy < 2; ++dy) {
                const int yy = y0 + dy;
                if ((unsigned)yy >= (unsigned)DIMN) continue;
                const float wzy = wz * (dy ? th : sh);
                #pragma unroll
                for (int dx = 0; dx < 2; ++dx) {
                    const int xx = x0 + dx;
                    if ((unsigned)xx >= (unsigned)DIMN) continue;
                    const float wc = wzy * (dx ? tw : sw);
                    const v2f v = *reinterpret_cast<const v2f*>(
                        imgB + (((size_t)((zz * DIMN + yy) * DIMN + xx)) << 1));
                    c0 += wc * v.x;
                    c1 += wc * v.y;
                }
            }
        }
        // streamed output: non-temporal so the image working set stays in L2
        v2f ov;
        ov.x = c0;
        ov.y = c1;
        __builtin_nontemporal_store(ov, reinterpret_cast<v2f*>(oImg) + idx);

        // nearest (order=0): jnp.round == round-half-even == rintf
        const int rz = (int)rintf(zd), ry = (int)rintf(zh), rx = (int)rintf(zw);
        float lv = 0.0f;
        if ((unsigned)rz < (unsigned)DIMN && (unsigned)ry < (unsigned)DIMN &&
            (unsigned)rx < (unsigned)DIMN) {
            lv = __builtin_nontemporal_load(lblB + (rz * DIMN + ry) * DIMN + rx);
        }
        __builtin_nontemporal_store(lv, oLbl + idx);
    }
}

// ---------------------------------------------------------------------------
extern "C" void kernel_launch(void* const* d_in, const int* in_sizes, int n_in,
                              void* d_out, int out_size, void* d_ws, size_t ws_size,
                              hipStream_t stream) {
    (void)in_sizes; (void)n_in; (void)out_size; (void)ws_size;

    const float* image  = (const float*)d_in[0];   // (2,128,128,128,2)
    const float* label  = (const float*)d_in[1];   // (2,128,128,128,1)
    const float* coarse = (const float*)d_in[2];   // (2,4,4,4,3)

    float* Kmat = (float*)d_ws;                    // 512 floats
    float* out_img = (float*)d_out;                // 8,388,608 floats
    float* out_lbl = out_img + (size_t)2 * DIMN * DIMN * DIMN * 2;

    build_K_kernel<<<1, DIMN, 0, stream>>>(Kmat);

    const int nblocks = 2 * DIMN * 2;              // one per (b, d, h-half)
    const size_t smem = (size_t)SMEM_FLOATS * sizeof(float);  // ~107 KB LDS
    elastic_warp_kernel<<<nblocks, NTHREADS, smem, stream>>>(
        image, label, coarse, Kmat, out_img, out_lbl);
}